// SparseCNN3D_81355270521213
// MI455X (gfx1250) — compile-verified
//
#include <hip/hip_runtime.h>

// ---------------------------------------------------------------------------
// Fused sparse-3D-CNN (depth-wise 1D conv stack) for gfx1250.
//   kernel 1: zero dense grid + mask (workspace)
//   kernel 2: scatter voxel features (atomicAdd) + mask set
//   kernel 3: fully fused 5-layer conv pipeline, one 16-wide HW tile per
//             block, all depths resident in LDS, fp32 WMMA 16x16x4.
// Round-2 changes: m-stride 65 -> 68 (conflict-free ds_load_b64 A-frags),
//                  dual accumulators to break the 48-deep WMMA chain.
// ---------------------------------------------------------------------------

typedef __attribute__((ext_vector_type(2))) float v2f;
typedef __attribute__((ext_vector_type(8))) float v8f;

#define BATCH   2
#define DEPTH   41          // D + 1 (D=40 even -> +1)
#define HW      16384       // 128*128
#define TILES   (HW / 16)   // 1024 tiles per batch
#define MST     68          // LDS m-stride: 8B-aligned pairs, conflict-free b64
#define RST     (16 * MST)  // LDS row stride = 1088 floats

// LDS layout (floats)
#define OFF_IN  0                       // padded dense: 43 rows * 16 m * 4 ch
#define OFF_M0  (OFF_IN + 43 * 64)      // mask0: 41*16
#define OFF_M1  (OFF_M0 + 41 * 16)      // mask1: 20*16
#define OFF_M2  (OFF_M1 + 20 * 16)      // mask2:  9*16
#define OFF_R1  (OFF_M2 + 9 * 16)       // region1: h1 (41 rows) / h3 (22 rows padded)
#define OFF_RA  (OFF_R1 + 41 * RST)     // regionA: h2 (22 rows padded) / h4 (20 rows)
#define SMEM_FLOATS (OFF_RA + 22 * RST) // = 72416 floats = 289664 bytes (283 KB)

// ---------------------------------------------------------------------------
__global__ void zero_ws_kernel(float* p, size_t n) {
    size_t i = (size_t)blockIdx.x * blockDim.x + threadIdx.x;
    size_t stride = (size_t)gridDim.x * blockDim.x;
    for (; i < n; i += stride) p[i] = 0.0f;
}

__global__ void scatter_kernel(const float* __restrict__ feats,
                               const int* __restrict__ coors,
                               float* __restrict__ dense,
                               float* __restrict__ maskg, int n) {
    int i = blockIdx.x * blockDim.x + threadIdx.x;
    if (i >= n) return;
    int b = coors[i * 4 + 0];
    int z = coors[i * 4 + 1];
    int y = coors[i * 4 + 2];
    int x = coors[i * 4 + 3];
    int cell = (b * DEPTH + z) * HW + y * 128 + x;
    atomicAdd(&dense[cell * 4 + 0], feats[i * 4 + 0]);
    atomicAdd(&dense[cell * 4 + 1], feats[i * 4 + 1]);
    atomicAdd(&dense[cell * 4 + 2], feats[i * 4 + 2]);
    atomicAdd(&dense[cell * 4 + 3], feats[i * 4 + 3]);
    maskg[cell] = 1.0f;
}

// ---------------------------------------------------------------------------
// One conv layer as GEMM tiles: M=16 spatial, N=16 out-ch per wave-tile,
// K = 3*CIN accumulated with V_WMMA_F32_16X16X4_F32 into two accumulators.
// wave -> ntile = wave&3 (out-ch group), depth group g = wave>>2.
// B fragments (weights) preloaded into registers, reused over all depths.
template <int CIN, bool TOGLOBAL>
__device__ __forceinline__ void conv_layer(
    const float* __restrict__ s_src, int srcMStride, int srcRowStride, int dstep,
    const float* __restrict__ wg,   // weights [3][CIN][64]
    const float* __restrict__ bg,   // bias [64]
    const float* __restrict__ s_mask, // [Dout][16]
    int Dout,
    float* dstL, int dstRowOff,     // LDS dest (if !TOGLOBAL), row stride RST
    float* __restrict__ outG, int b, int tileBase) {
    const int lane = threadIdx.x & 31;
    const int wave = threadIdx.x >> 5;
    const int n    = wave & 3;
    const int g    = wave >> 2;
    const int m    = lane & 15;
    const int hi   = lane >> 4;
    const int kk   = hi * 2;
    const int co   = n * 16 + m;    // lane&15 doubles as the N column

    constexpr int KBT = CIN / 4;    // k-blocks per tap
    constexpr int KB  = 3 * KBT;    // total k-blocks (K = 3*CIN)

    v2f bf[KB];
#pragma unroll
    for (int kb = 0; kb < KB; ++kb) {
        int tap = kb / KBT;
        int cib = (kb % KBT) * 4;
        bf[kb].x = wg[(tap * CIN + cib + kk)     * 64 + co];
        bf[kb].y = wg[(tap * CIN + cib + kk + 1) * 64 + co];
    }
    const float bias = bg[co];

    for (int d = g; d < Dout; d += 2) {
        v8f accs[2];
        accs[0] = (v8f){0.f, 0.f, 0.f, 0.f, 0.f, 0.f, 0.f, 0.f};
        accs[1] = (v8f){0.f, 0.f, 0.f, 0.f, 0.f, 0.f, 0.f, 0.f};
#pragma unroll
        for (int kb = 0; kb < KB; ++kb) {
            int tap = kb / KBT;
            int cib = (kb % KBT) * 4;
            const float* row = s_src + (d * dstep + tap) * srcRowStride;
            // two consecutive channels, 8B aligned -> single ds_load_b64
            v2f a = *(const v2f*)(row + m * srcMStride + cib + kk);
            accs[kb & 1] = __builtin_amdgcn_wmma_f32_16x16x4_f32(
                false, a, false, bf[kb], (short)0, accs[kb & 1], false, false);
        }
        v8f acc = accs[0] + accs[1];
#pragma unroll
        for (int r = 0; r < 8; ++r) {
            int Mr = r + 8 * hi;
            float mv = s_mask[d * 16 + Mr];
            float t  = (acc[r] + bias) * mv;
            t = t >= 0.0f ? t : 0.01f * t;   // leaky relu
            if constexpr (TOGLOBAL) {
                outG[((b * 64 + co) * 9 + d) * HW + tileBase + Mr] = t;
            } else {
                dstL[(d + dstRowOff) * RST + Mr * MST + co] = t;
            }
        }
    }
}

// ---------------------------------------------------------------------------
__global__ void __launch_bounds__(256, 1)
sparse_cnn_fused(const float* __restrict__ dense,
                 const float* __restrict__ maskg,
                 const float* __restrict__ w1, const float* __restrict__ b1,
                 const float* __restrict__ w2, const float* __restrict__ b2,
                 const float* __restrict__ w3, const float* __restrict__ b3,
                 const float* __restrict__ w4, const float* __restrict__ b4,
                 const float* __restrict__ w5, const float* __restrict__ b5,
                 float* __restrict__ out) {
    extern __shared__ float smem[];
    float* s_in = smem + OFF_IN;   // padded dense, 43 rows (row r = depth r-1)
    float* s_m0 = smem + OFF_M0;
    float* s_m1 = smem + OFF_M1;
    float* s_m2 = smem + OFF_M2;
    float* r1   = smem + OFF_R1;   // h1 (rows 0..40) then h3 (padded 0..21)
    float* rA   = smem + OFF_RA;   // h2 (padded 0..21) then h4 (rows 0..19)

    const int blk      = blockIdx.x;
    const int b        = blk >> 10;           // / TILES
    const int tileBase = (blk & (TILES - 1)) * 16;
    const int t        = threadIdx.x;

    // ---- init: load dense tile + mask, build downsampled masks, zero pads
    if (t < 64) { s_in[t] = 0.0f; s_in[42 * 64 + t] = 0.0f; }   // D pad rows
    for (int i = t; i < DEPTH * 64; i += 256) {                 // 41*16m*4c
        int d = i >> 6, rem = i & 63;                            // rem = m*4+c
        s_in[(d + 1) * 64 + rem] =
            dense[(((b * DEPTH + d) * HW + tileBase + (rem >> 2)) << 2) + (rem & 3)];
    }
    for (int i = t; i < DEPTH * 16; i += 256)
        s_m0[i] = maskg[(b * DEPTH + (i >> 4)) * HW + tileBase + (i & 15)];
    for (int i = t; i < RST; i += 256) {       // zero h2 pad rows 0 and 21
        rA[i] = 0.0f;
        rA[21 * RST + i] = 0.0f;
    }
    __syncthreads();
    for (int i = t; i < 20 * 16; i += 256) {   // mask downsample 41 -> 20
        int d = i >> 4, mm = i & 15;
        s_m1[i] = fmaxf(s_m0[(2 * d) * 16 + mm],
                        fmaxf(s_m0[(2 * d + 1) * 16 + mm],
                              s_m0[(2 * d + 2) * 16 + mm]));
    }
    __syncthreads();
    for (int i = t; i < 9 * 16; i += 256) {    // mask downsample 20 -> 9
        int d = i >> 4, mm = i & 15;
        s_m2[i] = fmaxf(s_m1[(2 * d) * 16 + mm],
                        fmaxf(s_m1[(2 * d + 1) * 16 + mm],
                              s_m1[(2 * d + 2) * 16 + mm]));
    }
    __syncthreads();

    // L1: C=4, 'same' over padded s_in, Dout=41 -> r1 rows 0..40
    conv_layer<4, false>(s_in, 4, 64, 1, w1, b1, s_m0, 41, r1, 0, nullptr, 0, 0);
    __syncthreads();
    // L2: stride-2 VALID over h1, Dout=20 -> rA rows 1..20 (padded)
    conv_layer<64, false>(r1, MST, RST, 2, w2, b2, s_m1, 20, rA, 1, nullptr, 0, 0);
    __syncthreads();
    // h1 dead: zero h3 pad rows inside region1
    for (int i = t; i < RST; i += 256) {
        r1[i] = 0.0f;
        r1[21 * RST + i] = 0.0f;
    }
    __syncthreads();
    // L3: 'same' over padded h2, Dout=20 -> r1 rows 1..20 (padded)
    conv_layer<64, false>(rA, MST, RST, 1, w3, b3, s_m1, 20, r1, 1, nullptr, 0, 0);
    __syncthreads();
    // L4: 'same' over padded h3, Dout=20 -> rA rows 0..19 (h2 dead)
    conv_layer<64, false>(r1, MST, RST, 1, w4, b4, s_m1, 20, rA, 0, nullptr, 0, 0);
    __syncthreads();
    // L5: stride-2 VALID over h4, Dout=9 -> global out (B,64,9,128,128)
    conv_layer<64, true>(rA, MST, RST, 2, w5, b5, s_m2, 9, nullptr, 0, out, b, tileBase);
}

// ---------------------------------------------------------------------------
extern "C" void kernel_launch(void* const* d_in, const int* in_sizes, int n_in,
                              void* d_out, int out_size, void* d_ws, size_t ws_size,
                              hipStream_t stream) {
    const float* feats = (const float*)d_in[0];
    const int*   coors = (const int*)d_in[1];
    const float* w1 = (const float*)d_in[3];
    const float* b1 = (const float*)d_in[4];
    const float* w2 = (const float*)d_in[5];
    const float* b2 = (const float*)d_in[6];
    const float* w3 = (const float*)d_in[7];
    const float* b3 = (const float*)d_in[8];
    const float* w4 = (const float*)d_in[9];
    const float* b4 = (const float*)d_in[10];
    const float* w5 = (const float*)d_in[11];
    const float* b5 = (const float*)d_in[12];
    float* out = (float*)d_out;

    float* dense = (float*)d_ws;                              // B*41*HW*4 f32
    float* maskg = dense + (size_t)BATCH * DEPTH * HW * 4;    // B*41*HW f32
    size_t zeroN = (size_t)BATCH * DEPTH * HW * 5;            // dense + mask
    int nvox = in_sizes[0] / 4;

    zero_ws_kernel<<<4096, 256, 0, stream>>>(dense, zeroN);
    scatter_kernel<<<(nvox + 255) / 256, 256, 0, stream>>>(feats, coors, dense,
                                                           maskg, nvox);

    size_t shmem = (size_t)SMEM_FLOATS * sizeof(float);       // 289664 B
    hipFuncSetAttribute((const void*)sparse_cnn_fused,
                        hipFuncAttributeMaxDynamicSharedMemorySize, (int)shmem);
    sparse_cnn_fused<<<BATCH * TILES, 256, shmem, stream>>>(
        dense, maskg, w1, b1, w2, b2, w3, b3, w4, b4, w5, b5, out);
    (void)n_in; (void)out_size; (void)ws_size;
}